// controller_68358699483504
// MI455X (gfx1250) — compile-verified
//
#include <hip/hip_runtime.h>

// ---------------------------------------------------------------------------
// Tiny sequential LSTM scan (L=1000 steps, HID=3, 5 cycling layer types) with
// the per-step 12x4 affine transform done by one V_WMMA_F32_16X16X4_F32:
//   gates = [W_hh | W_ih] @ [h0 h1 h2 x]^T + b
// A (16x4): every row = [h0 h1 h2 x]  -> D has identical rows -> lane L holds
// gates[L%16] in D[0] (ISA 7.12.2 C/D layout).
// B (4x16): B[k][n] = Wcat[n][k]; assumed VGPR layout mirrors the documented
// A-matrix 16x4 layout with N in place of M:
//   B vgpr0: lanes 0-15 K=0, lanes 16-31 K=2 ; B vgpr1: K=1 / K=3
// (compile-only environment; layout assumption documented here).
// This is a latency-bound serial problem: one wave32, all weights in VGPRs,
// native v_exp/v_rcp transcendentals, async predicated stores, parallel
// softmax epilogue kernel.
// ---------------------------------------------------------------------------

#define HID 3
#define NG 12   // 4*HID
#define NU 5    // N_UNIQUE layer types

typedef float v2f __attribute__((ext_vector_type(2)));
typedef float v8f __attribute__((ext_vector_type(8)));

__device__ __forceinline__ float fast_rcp(float x)  { return __builtin_amdgcn_rcpf(x); }
__device__ __forceinline__ float fast_exp2(float x) { return __builtin_amdgcn_exp2f(x); }

#define LOG2E 1.4426950408889634f

__global__ __launch_bounds__(32) void lstm_scan_wmma(
    const float* __restrict__ state,
    const float* __restrict__ W_ih,   // (5, 12, 1)
    const float* __restrict__ W_hh,   // (5, 12, 3)
    const float* __restrict__ b_ih,   // (5, 12)
    const float* __restrict__ b_hh,   // (5, 12)
    float* __restrict__ out,          // (L, 3) raw h (softmax applied later)
    int L)
{
  const int  lane    = threadIdx.x;     // 0..31 (wave32)
  const int  n       = lane & 15;       // gate / B-column index
  const bool hiHalf  = lane >= 16;      // holds K=2,3 slices of A/B
  const bool isTanh  = (n >= 6) && (n < 9);   // g-gate lanes

  // ---- per-layer-type constants resident in VGPRs ----
  float B0[NU], B1[NU], bv[NU];
#pragma unroll
  for (int l = 0; l < NU; ++l) {
    float w0 = 0.f, w1 = 0.f, bb = 0.f;
    if (n < NG) {
      const float* wh = W_hh + (l * NG + n) * HID;
      if (!hiHalf) { w0 = wh[0];  w1 = wh[1]; }                 // K=0, K=1
      else         { w0 = wh[2];  w1 = W_ih[l * NG + n]; }      // K=2, K=3(=wi)
      bb = b_ih[l * NG + n] + b_hh[l * NG + n];
    }
    B0[l] = w0; B1[l] = w1; bv[l] = bb;
  }

  float h = 0.f, c = 0.f;               // valid in lanes 0..2
  int   base    = 0;
  float xs_next = (base + lane < L) ? state[base + lane] : 0.f;  // chunk prefetch
  float xs_cur  = 0.f;

  for (int t0 = 0; t0 < L; t0 += NU) {
#pragma unroll
    for (int l = 0; l < NU; ++l) {
      const int t = t0 + l;
      if (t < L) {                                 // uniform guard
        if ((t & 31) == 0) {                       // rotate x chunk + prefetch next
          xs_cur = xs_next;
          base  += 32;
          xs_next = (base + lane < L) ? state[base + lane] : 0.f;
        }
        const float x = __shfl(xs_cur, t & 31, 32);

        // Build A (rows identical = [h0 h1 h2 x]):
        //  vgpr0 : K=0 (h0) in lanes 0-15, K=2 (h2) in lanes 16-31
        //  vgpr1 : K=1 (h1) in lanes 0-15, K=3 (x)  in lanes 16-31
        const float a0  = __shfl(h, hiHalf ? 2 : 0, 32);
        const float h1b = __shfl(h, 1, 32);
        const float a1  = hiHalf ? x : h1b;

        v2f A; A.x = a0;    A.y = a1;
        v2f B; B.x = B0[l]; B.y = B1[l];
        v8f C = {};
        // 8-arg form: (neg_a, A, neg_b, B, c_mod, C, reuse_a, reuse_b)
        v8f D = __builtin_amdgcn_wmma_f32_16x16x4_f32(
                    false, A, false, B, (short)0, C, false, false);

        const float gate = D[0] + bv[l];           // lane L: gates[L%16]

        // act = sigmoid(gate), or tanh(gate)=2*sigmoid(2*gate)-1 on g lanes
        const float tin = isTanh ? (gate + gate) : gate;
        const float sg  = fast_rcp(1.f + fast_exp2(-tin * LOG2E));
        const float act = isTanh ? (sg + sg - 1.f) : sg;

        // lane k<3 gathers i=act[k] (own), f=act[k+3], g=act[k+6], o=act[k+9]
        const float fv = __shfl(act, lane + 3, 32);
        const float gv = __shfl(act, lane + 6, 32);
        const float ov = __shfl(act, lane + 9, 32);

        c = fv * c + act * gv;
        const float e2 = fast_exp2(2.f * c * LOG2E);        // tanh(c)
        const float th = 1.f - 2.f * fast_rcp(e2 + 1.f);
        h = ov * th;

        if (lane < HID) out[t * HID + lane] = h;   // fire-and-forget store
      }
    }
  }
}

// Independent, fully parallel epilogue: in-place row softmax over (L, 3).
__global__ void softmax_rows(float* __restrict__ out, int rows)
{
  const int r = blockIdx.x * blockDim.x + threadIdx.x;
  if (r >= rows) return;
  float* p = out + r * HID;
  const float a = p[0], b = p[1], cc = p[2];
  const float m  = fmaxf(a, fmaxf(b, cc));
  const float ea = fast_exp2((a  - m) * LOG2E);
  const float eb = fast_exp2((b  - m) * LOG2E);
  const float ec = fast_exp2((cc - m) * LOG2E);
  const float inv = fast_rcp(ea + eb + ec);
  p[0] = ea * inv; p[1] = eb * inv; p[2] = ec * inv;
}

extern "C" void kernel_launch(void* const* d_in, const int* in_sizes, int n_in,
                              void* d_out, int out_size, void* d_ws, size_t ws_size,
                              hipStream_t stream)
{
  (void)n_in; (void)out_size; (void)d_ws; (void)ws_size;
  const float* state = (const float*)d_in[0];
  const float* W_ih  = (const float*)d_in[1];
  const float* W_hh  = (const float*)d_in[2];
  const float* b_ih  = (const float*)d_in[3];
  const float* b_hh  = (const float*)d_in[4];
  float* out = (float*)d_out;
  const int L = in_sizes[0];             // 1000 (multiple of NU=5)

  lstm_scan_wmma<<<1, 32, 0, stream>>>(state, W_ih, W_hh, b_ih, b_hh, out, L);
  softmax_rows<<<(L + 255) / 256, 256, 0, stream>>>(out, L);
}